// LIF_46462956208158
// MI455X (gfx1250) — compile-verified
//
#include <hip/hip_runtime.h>

// LIF forward scan: mem = mem*TAU + x_t; spk = (mem >= thresh); mem = spk ? 0 : mem
// x: [B, T, N] f32, thresh: [N] f32, out spikes: [B, T, N] f32.
//
// Roofline: 26.2M elements, read 105MB + write 105MB = 210MB @ 23.3 TB/s -> ~9us.
// Elementwise sequential scan over T: no contraction dim, so WMMA is structurally
// inapplicable. Optimal shape = B128 coalesced non-temporal streams + deep
// global_prefetch_b8 pipelining (prefetch doesn't consume LOADcnt -> unbounded
// depth into GL2) + 2D grid to avoid any integer division.

typedef float v4f __attribute__((ext_vector_type(4)));

#define LIF_TAU 0.5f
#define PF_DIST 8   // timesteps of prefetch lead: ~8 * 90ns loop time > HBM latency

__global__ __launch_bounds__(256) void lif_fwd_kernel(
    const float* __restrict__ x,
    const float* __restrict__ thresh,
    float* __restrict__ out,
    int nv,   // N / 4 (vec4 columns)
    int T)
{
    const int c = blockIdx.x * 256 + threadIdx.x;  // vec4 column
    if (c >= nv) return;
    const int b = blockIdx.y;                      // batch index (no division)

    const v4f th = ((const v4f*)thresh)[c];

    const size_t stride = (size_t)nv;              // vec4 elems between timesteps
    const size_t base   = ((size_t)b * (size_t)T) * stride + (size_t)c;
    const v4f* __restrict__ xp = (const v4f*)x + base;
    v4f* __restrict__ op       = (v4f*)out + base;

    v4f mem = (v4f)0.0f;

    #pragma unroll 2
    for (int t = 0; t < T; ++t) {
        // gfx1250 global_prefetch_b8, PF_DIST timesteps (128KB/thread-stride)
        // ahead. Address is independent of the serial mem-chain; prefetch is
        // not tracked by LOADcnt, so this gives arbitrarily deep MLP into GL2.
        if (t + PF_DIST < T) {
            __builtin_prefetch((const void*)(xp + (size_t)PF_DIST * stride), 0, 0);
        }

        // Streaming (non-temporal) B128 load: 512B contiguous per wave32.
        v4f xt = __builtin_nontemporal_load(xp);

        // Leaky integrate: mem = mem * TAU + x_t
        mem.x = mem.x * LIF_TAU + xt.x;
        mem.y = mem.y * LIF_TAU + xt.y;
        mem.z = mem.z * LIF_TAU + xt.z;
        mem.w = mem.w * LIF_TAU + xt.w;

        // Fire (Heaviside forward); compiler CSEs the compare for the reset.
        v4f spk;
        spk.x = (mem.x >= th.x) ? 1.0f : 0.0f;
        spk.y = (mem.y >= th.y) ? 1.0f : 0.0f;
        spk.z = (mem.z >= th.z) ? 1.0f : 0.0f;
        spk.w = (mem.w >= th.w) ? 1.0f : 0.0f;

        // Hard reset: (1 - spk) * mem  ==  select(spk, 0, mem)
        mem.x = (mem.x >= th.x) ? 0.0f : mem.x;
        mem.y = (mem.y >= th.y) ? 0.0f : mem.y;
        mem.z = (mem.z >= th.z) ? 0.0f : mem.z;
        mem.w = (mem.w >= th.w) ? 0.0f : mem.w;

        // Streaming B128 store of spikes.
        __builtin_nontemporal_store(spk, op);

        xp += stride;
        op += stride;
    }
}

extern "C" void kernel_launch(void* const* d_in, const int* in_sizes, int n_in,
                              void* d_out, int out_size, void* d_ws, size_t ws_size,
                              hipStream_t stream) {
    const float* x      = (const float*)d_in[0];   // [B, T, N] f32
    const float* thresh = (const float*)d_in[1];   // [N] f32
    float* out          = (float*)d_out;           // [B, T, N] f32

    const int N  = in_sizes[1];                    // 4096
    const long long total = (long long)in_sizes[0];
    const int T  = 100;                            // reference scan length
    const int BT = (int)(total / N);               // B*T = 6400
    const int B  = BT / T;                         // 64

    const int nv = N >> 2;                         // 1024 vec4 columns

    dim3 grid((nv + 255) / 256, B);                // 4 x 64 blocks, 8 wave32s each
    lif_fwd_kernel<<<grid, 256, 0, stream>>>(x, thresh, out, nv, T);
}